// DispMVS_64527588655384
// MI455X (gfx1250) — compile-verified
//
#include <hip/hip_runtime.h>
#include <cstdint>

// ---------------------------------------------------------------------------
// DispMVS depth fusion for gfx1250 (MI455X).
// Memory-bound: mask stream (189 MB) >> everything else -> ~8us floor @23.3TB/s.
// Strategy: async global->LDS staging (CDNA5 ASYNCcnt path) with NT hints,
// double-buffered, per-wave waits (no barriers: each lane owns its LDS slots).
// LDS layout [tap][thread][subpix] -> 128-bit ds drains; fused phases write one
// aligned float4 per thread -> fully coalesced b128 output stores.
// ---------------------------------------------------------------------------

#define EPSF 1e-6f
constexpr int NN  = 2;            // neighbors
constexpr int NB  = 2;            // batch
constexpr int H   = 256;
constexpr int W   = 320;
constexpr int HW  = H * W;
constexpr int UP  = 4;
constexpr int TPB = 160;          // 5 full wave32s; half a row per block
constexpr int CPP = 36;           // channels staged per phase (9 taps x 4 subpix)

__device__ __forceinline__ void inv3x3(const float m[9], float o[9]) {
    float a=m[0],b=m[1],c=m[2],d=m[3],e=m[4],f=m[5],g=m[6],h=m[7],i=m[8];
    float A  =  (e*i - f*h);
    float Bc = -(d*i - f*g);
    float C  =  (d*h - e*g);
    float det = a*A + b*Bc + c*C;
    float r = 1.0f / det;
    o[0]=A*r;  o[1]=-(b*i - c*h)*r; o[2]= (b*f - c*e)*r;
    o[3]=Bc*r; o[4]= (a*i - c*g)*r; o[5]=-(a*f - c*d)*r;
    o[6]=C*r;  o[7]=-(a*h - b*g)*r; o[8]= (a*e - b*d)*r;
}

// ---------------------------------------------------------------------------
// Pass 1: per-pixel geometry -> inv_d (NN,B,H,W) into workspace. ALU-trivial.
// ---------------------------------------------------------------------------
__global__ void geom_kernel(const float* __restrict__ Kref,
                            const float* __restrict__ Knei,
                            const float* __restrict__ Rnei,
                            const float* __restrict__ Tnei,
                            const float* __restrict__ depth0,
                            const float* __restrict__ flow,
                            const float* __restrict__ dminp,
                            const float* __restrict__ dmaxp,
                            float* __restrict__ invd) {
    int idx = blockIdx.x * blockDim.x + threadIdx.x;
    if (idx >= NN * NB * HW) return;
    int x = idx % W;
    int y = (idx / W) % H;
    int b = (idx / HW) % NB;
    int n = idx / (HW * NB);
    int nb = n * NB + b;

    float Kr[9], Kri[9];
#pragma unroll
    for (int i = 0; i < 9; i++) Kr[i] = Kref[b * 9 + i];
    inv3x3(Kr, Kri);
    // xyz = Kref^{-1} [x,y,1]; then normalize by z (z component -> 1)
    float X0 = Kri[0]*x + Kri[1]*y + Kri[2];
    float Y0 = Kri[3]*x + Kri[4]*y + Kri[5];
    float Z0 = Kri[6]*x + Kri[7]*y + Kri[8];
    float xz = X0 / Z0, yz = Y0 / Z0;      // z == 1

    float Kn[9], R[9], Kni[9], T3[3];
#pragma unroll
    for (int i = 0; i < 9; i++) { Kn[i] = Knei[nb*9+i]; R[i] = Rnei[nb*9+i]; }
#pragma unroll
    for (int i = 0; i < 3; i++) T3[i] = Tnei[nb*3+i];
    inv3x3(Kn, Kni);

    float d0 = depth0[b * HW + y * W + x];
    float fl = flow[nb * HW + y * W + x];
    float dmin = dminp[0], dmax = dmaxp[0];

    auto proj = [&](float d, float& px, float& py) {
        float vx = xz * d, vy = yz * d, vz = d;
        float rx = R[0]*vx + R[1]*vy + R[2]*vz + T3[0];
        float ry = R[3]*vx + R[4]*vy + R[5]*vz + T3[1];
        float rz = R[6]*vx + R[7]*vy + R[8]*vz + T3[2];
        float p0 = Kn[0]*rx + Kn[1]*ry + Kn[2]*rz;
        float p1 = Kn[3]*rx + Kn[4]*ry + Kn[5]*rz;
        float p2 = Kn[6]*rx + Kn[7]*ry + Kn[8]*rz;
        float iz = 1.0f / (fabsf(p2) + EPSF);
        px = p0 * iz; py = p1 * iz;
    };

    float sx, sy, ex, ey;
    proj(d0, sx, sy);
    proj(d0 * 2.0f + 10.0f, ex, ey);
    float fdx = ex - sx, fdy = ey - sy;
    float nrm = sqrtf(fdx*fdx + fdy*fdy);
    fdx /= nrm; fdy /= nrm;
    float mxp = sx + fdx * fl, myp = sy + fdy * fl;
    bool fm = fabsf(fdy) > fabsf(fdx);

    float rx = R[0]*xz + R[1]*yz + R[2];
    float ry = R[3]*xz + R[4]*yz + R[5];
    float rz = R[6]*xz + R[7]*yz + R[8];

    float nx0 = Kni[0]*mxp + Kni[1]*myp + Kni[2];
    float ny0 = Kni[3]*mxp + Kni[4]*myp + Kni[5];
    float nz0 = Kni[6]*mxp + Kni[7]*myp + Kni[8];
    float izn = 1.0f / (fabsf(nz0) + EPSF);
    float nx = nx0 * izn, ny = ny0 * izn;

    float tx = T3[0], ty = T3[1], tz = T3[2];
    float ivx = fabsf(rx - rz*nx) / (fabsf(tz*nx - tx) + EPSF);
    float ivy = fabsf(ry - rz*ny) / (fabsf(tz*ny - ty) + EPSF);
    float iv = fm ? ivy : ivx;
    iv = (iv - dmax) / (dmin - dmax);
    iv = fminf(fmaxf(iv, 0.0f), 1.0f);
    invd[idx] = iv;
}

// ---------------------------------------------------------------------------
// Pass 2: convex upsample + 2-way fusion. Streams the 144-ch mask via
// global_load_async_to_lds_b32 (NT), double-buffered, per-wave ASYNCcnt waits.
// Block = (b, y, half-row of 160 px); thread = one low-res pixel.
// ---------------------------------------------------------------------------
__global__ __launch_bounds__(TPB) void fuse_kernel(const float* __restrict__ invd,
                                                   const float* __restrict__ conf,
                                                   const float* __restrict__ mask,
                                                   const float* __restrict__ dminp,
                                                   const float* __restrict__ dmaxp,
                                                   float* __restrict__ out) {
    // Layout: [buffer][tap k][thread][subpix jj]  -> 16B/thread/tap chunks,
    // drained with ds_load_b128. 2 x 23KB.
    __shared__ alignas(16) float smask[2][9 * TPB * 4];

    const int t     = threadIdx.x;
    const int bid   = blockIdx.x;
    const int chunk = bid & 1;              // which half of the row
    const int y     = (bid >> 1) % H;
    const int b     = bid / (2 * H);
    const int x     = chunk * TPB + t;

    // Phase p (0..7): jg = p>>1 selects sub-pixel row (4 columns jj), n = p&1 neighbor.
    // Channel c = k*16 + jg*4 + jj  (k = tap 0..8, jj = 0..3).
    auto stage = [&](int bi, int p) {
        const int jg = p >> 1, n = p & 1;
        const float* mb = mask + ((size_t)(n * NB + b) * 144) * HW
                               + (size_t)y * W + x;
        uint32_t lbase = (uint32_t)(uintptr_t)(&smask[bi][0]) + (uint32_t)(t * 16);
#pragma unroll
        for (int k = 0; k < 9; k++) {
#pragma unroll
            for (int jj = 0; jj < 4; jj++) {
                const int c = k * 16 + jg * 4 + jj;
                const float* g = mb + (size_t)c * HW;
                uint32_t l = lbase + (uint32_t)(k * TPB * 16 + jj * 4);
                // CDNA5 async global->LDS, non-temporal (mask is stream-once)
                asm volatile("global_load_async_to_lds_b32 %0, %1, off th:TH_LOAD_NT"
                             :: "v"(l), "v"(g) : "memory");
            }
        }
    };

    stage(0, 0);    // kick the pipeline before doing the small register loads

    // 3x3 zero-padded neighborhoods of inv_d and conf for both neighbors.
    float u_inv[2][9], u_conf[2][9];
#pragma unroll
    for (int n = 0; n < 2; n++) {
#pragma unroll
        for (int k = 0; k < 9; k++) {
            const int dy = k / 3 - 1, dx = k % 3 - 1;
            const int yy = y + dy, xx = x + dx;
            const bool ok = (yy >= 0 && yy < H && xx >= 0 && xx < W);
            const size_t off = (size_t)(n * NB + b) * HW + (size_t)yy * W + xx;
            u_inv[n][k]  = ok ? invd[off] : 0.0f;
            u_conf[n][k] = ok ? conf[off] : 0.0f;
        }
    }

    const float dmin = dminp[0], dmax = dmaxp[0];
    float acc_i[4], acc_c[4];
    float* outb = out + (size_t)b * (UP * H) * (UP * W);

    for (int p = 0; p < 8; p++) {
        const int jg = p >> 1, n = p & 1;
        if (p < 7) {
            stage((p + 1) & 1, p + 1);
            // async completions are in-order: <=36 outstanding means the
            // current buffer's 36 loads have all landed in LDS.
            asm volatile("s_wait_asynccnt 36" ::: "memory");
        } else {
            asm volatile("s_wait_asynccnt 0" ::: "memory");
        }
        const float* buf = &smask[p & 1][0];

        // Drain this thread's 9 taps x 4 subpix as nine 128-bit LDS loads.
        float vv[9][4];
#pragma unroll
        for (int k = 0; k < 9; k++) {
            const float4 q = *(const float4*)&buf[(k * TPB + t) * 4];
            vv[k][0] = q.x; vv[k][1] = q.y; vv[k][2] = q.z; vv[k][3] = q.w;
        }

        float4 res;
        float* resf = (float*)&res;
#pragma unroll
        for (int jj = 0; jj < 4; jj++) {
            float mx = -1e30f;
#pragma unroll
            for (int k = 0; k < 9; k++) mx = fmaxf(mx, vv[k][jj]);
            float se = 0.0f, si = 0.0f, sc = 0.0f;
#pragma unroll
            for (int k = 0; k < 9; k++) {
                float e = __expf(vv[k][jj] - mx);
                se += e;
                si += e * u_inv[n][k];
                sc += e * u_conf[n][k];
            }
            float r  = 1.0f / se;
            float iu = si * r;       // inv_up at this sub-pixel
            float cu = sc * r;       // conf_up at this sub-pixel

            if (n == 0) {
                acc_i[jj] = iu;
                acc_c[jj] = cu;
            } else {
                // 2-way softmax fusion over neighbors + final reciprocal
                float c0 = acc_c[jj], c1 = cu;
                float m  = fmaxf(c0, c1);
                float e0 = __expf(c0 - m), e1 = __expf(c1 - m);
                float w0 = e0 / (e0 + e1);
                float df = acc_i[jj] * w0 + iu * (1.0f - w0);
                df = df * (dmin - dmax) + dmax;
                resf[jj] = 1.0f / df;
            }
        }

        if (n == 1) {
            // All 4 sub-pixels of this phase share row Y = y*4+jg and occupy
            // consecutive X = x*4..x*4+3 -> one aligned, coalesced b128 store.
            *(float4*)&outb[(size_t)(y * UP + jg) * (UP * W) + x * UP] = res;
        }
    }
}

// ---------------------------------------------------------------------------
extern "C" void kernel_launch(void* const* d_in, const int* in_sizes, int n_in,
                              void* d_out, int out_size, void* d_ws, size_t ws_size,
                              hipStream_t stream) {
    const float* Kref   = (const float*)d_in[0];
    const float* Knei   = (const float*)d_in[1];
    const float* Rnei   = (const float*)d_in[2];
    const float* Tnei   = (const float*)d_in[3];
    const float* depth0 = (const float*)d_in[4];
    const float* flow   = (const float*)d_in[5];
    const float* mask   = (const float*)d_in[6];
    const float* conf   = (const float*)d_in[7];
    const float* dmin   = (const float*)d_in[8];
    const float* dmax   = (const float*)d_in[9];
    float* out  = (float*)d_out;
    float* invd = (float*)d_ws;            // NN*B*H*W floats = 1.31 MB scratch

    const int total = NN * NB * HW;
    geom_kernel<<<(total + 255) / 256, 256, 0, stream>>>(
        Kref, Knei, Rnei, Tnei, depth0, flow, dmin, dmax, invd);

    fuse_kernel<<<NB * H * (W / TPB), TPB, 0, stream>>>(
        invd, conf, mask, dmin, dmax, out);
}